// Poisson_28028956573752
// MI455X (gfx1250) — compile-verified
//
#include <hip/hip_runtime.h>
#include <hip/hip_bf16.h>

typedef __attribute__((ext_vector_type(2))) float v2f;
typedef __attribute__((ext_vector_type(8))) float v8f;

#define PI_F 3.14159265358979323846f
#define TWO_PI2_F 19.739208802178716f /* 2*pi^2 */

// ---------------------------------------------------------------------------
// Kernel A: per-triangle quadrature. Computes s[t] = area_t * K_t (to ws) and
// scatter-adds the load vector b (hardware global_atomic_add_f32).
// Streaming, memory-bound: elem (b96), 3x node gather (b64, L2-resident),
// area (b32), s store (b32), 3 atomics.
// ---------------------------------------------------------------------------
__global__ void __launch_bounds__(256)
fem_coeff_kernel(const float* __restrict__ node,
                 const int*   __restrict__ elem,
                 const float* __restrict__ area,
                 float* __restrict__ s,
                 float* __restrict__ b,
                 int NT)
{
    int t = blockIdx.x * blockDim.x + threadIdx.x;
    if (t >= NT) return;

    int i0 = elem[3 * t + 0];
    int i1 = elem[3 * t + 1];
    int i2 = elem[3 * t + 2];

    float x0 = node[2 * (long)i0], y0 = node[2 * (long)i0 + 1];
    float x1 = node[2 * (long)i1], y1 = node[2 * (long)i1 + 1];
    float x2 = node[2 * (long)i2], y2 = node[2 * (long)i2 + 1];
    float a  = area[t];

    // order-2 quad pts: PHI rows (2/3,1/6,1/6) rotated; W = 1/3 each.
    const float lc = 2.0f / 3.0f, ls = 1.0f / 6.0f;
    float f0, f1, f2, Ksum = 0.0f;
    {
        float px = lc * x0 + ls * x1 + ls * x2;
        float py = lc * y0 + ls * y1 + ls * y2;
        float sx = __sinf(PI_F * px), sy = __sinf(PI_F * py), cy = __cosf(PI_F * py);
        Ksum += 1.0f + 0.1f * sx * cy;
        f0 = TWO_PI2_F * sx * sy;
    }
    {
        float px = ls * x0 + lc * x1 + ls * x2;
        float py = ls * y0 + lc * y1 + ls * y2;
        float sx = __sinf(PI_F * px), sy = __sinf(PI_F * py), cy = __cosf(PI_F * py);
        Ksum += 1.0f + 0.1f * sx * cy;
        f1 = TWO_PI2_F * sx * sy;
    }
    {
        float px = ls * x0 + ls * x1 + lc * x2;
        float py = ls * y0 + ls * y1 + lc * y2;
        float sx = __sinf(PI_F * px), sy = __sinf(PI_F * py), cy = __cosf(PI_F * py);
        Ksum += 1.0f + 0.1f * sx * cy;
        f2 = TWO_PI2_F * sx * sy;
    }

    float K = Ksum * (1.0f / 3.0f);
    s[t] = a * K;

    // bt_i = area * sum_p W[p]*PHI[p,i]*f_p = (area/3) * (fsum/6 + f_i/2)
    float fsum  = f0 + f1 + f2;
    float third = a * (1.0f / 3.0f);
    float base  = fsum * (1.0f / 6.0f);
    atomicAdd(&b[i0], third * (base + 0.5f * f0));
    atomicAdd(&b[i1], third * (base + 0.5f * f1));
    atomicAdd(&b[i2], third * (base + 0.5f * f2));
}

// ---------------------------------------------------------------------------
// Kernel B: batched 3x3 Gram matrices via V_WMMA_F32_16X16X4_F32.
// One wave handles 5 triangles per WMMA: rows/cols 3q+i (q=0..4), K dim holds
// Dlambda's d=0,1 (K=2,3 zero-padded -> lanes 16..31 of A hold zeros).
// ISA layout trick: the A-matrix (16x4) register image equals the B-matrix
// (4x16) register image for this symmetric Gram product, so a single v2f per
// lane feeds both operands. Diagonal 3x3 blocks of D are extracted, scaled by
// s[t]=area*K, and stored to Avals.
// ---------------------------------------------------------------------------
__global__ void __launch_bounds__(256)
fem_gram_wmma_kernel(const float* __restrict__ Dl,
                     const float* __restrict__ s,
                     float* __restrict__ Avals,
                     int NT, int ngroups)
{
    int wave = (int)((blockIdx.x * blockDim.x + threadIdx.x) >> 5);
    int lane = threadIdx.x & 31;
    if (wave >= ngroups) return;          // wave-uniform: EXEC stays all-ones

    long tbase = (long)wave * 5;

    // Build A (== B). Lanes 0..14: lane -> (q = lane/3, i = lane%3).
    //   A vgpr0 lane L (<16) = A[M=L, K=0] = Dlambda[t, d=0, i]
    //   A vgpr1 lane L (<16) = A[M=L, K=1] = Dlambda[t, d=1, i]
    //   lanes 16..31 = K=2,3 -> zero (K padded from 2 to 4)
    v2f amat = {0.0f, 0.0f};
    if (lane < 15) {
        long t = tbase + lane / 3;
        if (t < NT) {
            int i = lane % 3;
            amat.x = Dl[t * 6 + 0 + i];   // Dlambda[t,0,i]
            amat.y = Dl[t * 6 + 3 + i];   // Dlambda[t,1,i]
        }
    }

    v8f c = {0.f, 0.f, 0.f, 0.f, 0.f, 0.f, 0.f, 0.f};
    // D[m,n] = sum_d Dl[t_m, d, i_m] * Dl[t_n, d, j_n]; diagonal blocks = Gram
    v8f d = __builtin_amdgcn_wmma_f32_16x16x4_f32(
        /*neg_a=*/false, amat, /*neg_b=*/false, amat,
        /*c_mod=*/(short)0, c, /*reuse_a=*/false, /*reuse_b=*/false);

    // Extract: lane L holds D[row, col] for col = L%16,
    //   rows 0..7 in lanes 0..15 (vgpr r), rows 8..15 in lanes 16..31.
    int col = lane & 15;
    if (col < 15) {
        int q = col / 3;
        int j = col - 3 * q;
        long t = tbase + q;
        if (t < NT) {
            float sc = s[t];              // area_t * K_t
            bool low_half = (lane < 16);
#pragma unroll
            for (int i = 0; i < 3; ++i) {
                int row = 3 * q + i;
                if (low_half == (row < 8)) {
                    Avals[t * 9 + i * 3 + j] = sc * d[row & 7];
                }
            }
        }
    }
}

// ---------------------------------------------------------------------------
// Launch: memset b region -> coeff kernel (s + b atomics) -> WMMA Gram kernel.
// Inputs: node[N,2] f32, elem[NT,3] i32, Dlambda[NT,2,3] f32, area[NT] f32,
//         isBdNode[N] bool (unused by reference outputs).
// d_out = Avals[NT*9] ++ b[N] (f32).
// ---------------------------------------------------------------------------
extern "C" void kernel_launch(void* const* d_in, const int* in_sizes, int n_in,
                              void* d_out, int out_size, void* d_ws, size_t ws_size,
                              hipStream_t stream) {
    const float* node = (const float*)d_in[0];
    const int*   elem = (const int*)d_in[1];
    const float* Dl   = (const float*)d_in[2];
    const float* area = (const float*)d_in[3];

    const int N  = in_sizes[0] / 2;
    const int NT = in_sizes[3];

    float* Avals = (float*)d_out;
    float* b     = Avals + (size_t)NT * 9;
    float* s     = (float*)d_ws;          // NT floats of scratch

    // b is accumulated with atomics; d_out is poisoned -> zero it first.
    hipMemsetAsync(b, 0, (size_t)N * sizeof(float), stream);

    {
        int threads = 256;
        int blocks  = (NT + threads - 1) / threads;
        fem_coeff_kernel<<<blocks, threads, 0, stream>>>(node, elem, area, s, b, NT);
    }
    {
        int ngroups = (NT + 4) / 5;       // 5 triangles per wave
        int threads = 256;                // 8 waves per block
        int wavesPerBlock = threads / 32;
        int blocks  = (ngroups + wavesPerBlock - 1) / wavesPerBlock;
        fem_gram_wmma_kernel<<<blocks, threads, 0, stream>>>(Dl, s, Avals, NT, ngroups);
    }
}